// SpatialAttentionModule_67843303407694
// MI455X (gfx1250) — compile-verified
//
#include <hip/hip_runtime.h>
#include <math.h>

#define NV   40962
#define NC   256
#define NBATCH 16
#define KNB  7
#define NVH  (NV / 2)                      // 20481 float2 per (b,c) row
#define TILES_PER_BATCH ((NV + 15) / 16)   // 2561

typedef __attribute__((ext_vector_type(2))) float v2f;
typedef __attribute__((ext_vector_type(8))) float v8f;

// ---------------------------------------------------------------------------
// Kernel 1: channel reduction (mean + max over 256 channels).
// Each thread owns 2 consecutive vertices (float2 loads, 256B/wave/issue,
// fully coalesced). Two independent sum/max chains per component hide VALU
// latency; traffic = one pass over x (671 MB) -> ~29us at 23.3 TB/s.
// ---------------------------------------------------------------------------
__global__ void __launch_bounds__(256)
pool_reduce_kernel(const float* __restrict__ x, float2* __restrict__ pool) {
    int i = blockIdx.x * blockDim.x + threadIdx.x;
    if (i >= NBATCH * NVH) return;
    int b  = i / NVH;
    int vh = i - b * NVH;                  // float2 index within batch
    const float2* p2 = (const float2*)(x + (size_t)b * NC * NV) + vh;

    float sx0 = 0.f, sy0 = 0.f, sx1 = 0.f, sy1 = 0.f;
    float mx0 = -__builtin_inff(), my0 = -__builtin_inff();
    float mx1 = -__builtin_inff(), my1 = -__builtin_inff();

    #pragma unroll 4
    for (int c = 0; c < NC; c += 2) {
        float2 q0 = p2[(size_t)(c + 0) * NVH];
        float2 q1 = p2[(size_t)(c + 1) * NVH];
        sx0 += q0.x;  sy0 += q0.y;
        sx1 += q1.x;  sy1 += q1.y;
        mx0 = fmaxf(mx0, q0.x);  my0 = fmaxf(my0, q0.y);
        mx1 = fmaxf(mx1, q1.x);  my1 = fmaxf(my1, q1.y);
    }
    const float inv = 1.0f / (float)NC;
    int v = vh * 2;
    pool[(size_t)b * NV + v]     = make_float2((sx0 + sx1) * inv, fmaxf(mx0, mx1));
    pool[(size_t)b * NV + v + 1] = make_float2((sy0 + sy1) * inv, fmaxf(my0, my1));
}

// ---------------------------------------------------------------------------
// Kernel 2: neighbor gather + einsum('bnf,of->bno') via V_WMMA_F32_16X16X4_F32
// + sigmoid. One wave per 16-vertex tile of one batch.
//
// A (16x4 f32): lanes 0-15 row M=lane supply K0/K1, lanes 16-31 supply K2/K3.
//   Per step s the K-slice is features 4s..4s+3 = float2 pairs g[2s], g[2s+1]:
//   low-half lanes take g[2s], high-half g[2s+1]  (constant indices -> no
//   runtime-indexed array, just one cndmask per component).
// B (4x16 f32): column 0 lives in lanes 0 (K0/K1) and 16 (K2/K3): lane 0 gets
//   W pair 2s, lane 16 W pair 2s+1, all other lanes 0.
// D column 0: lane 0 = rows 0..7 (VGPR 0..7), lane 16 = rows 8..15.
// ---------------------------------------------------------------------------
__global__ void __launch_bounds__(256)
attention_wmma_kernel(const float2* __restrict__ pool,
                      const int*    __restrict__ neigh,
                      const float*  __restrict__ W,
                      const float*  __restrict__ bias,
                      float*        __restrict__ out) {
    int gtid = blockIdx.x * blockDim.x + threadIdx.x;
    int wave = gtid >> 5;
    int lane = threadIdx.x & 31;
    if (wave >= NBATCH * TILES_PER_BATCH) return;   // wave-uniform: EXEC all-1s

    int batch = wave / TILES_PER_BATCH;
    int tile  = wave % TILES_PER_BATCH;
    int row   = lane & 15;                 // matrix row M for this lane
    bool hi   = (lane >> 4) != 0;          // false: K0/K1 lanes, true: K2/K3
    int v     = tile * 16 + row;
    int vc    = (v < NV) ? v : (NV - 1);   // clamp padded rows

    // Gather 7 neighbor (avg,max) pairs for this row; pair 7 = zero padding.
    float2 g[8];
    #pragma unroll
    for (int j = 0; j < KNB; ++j) {
        int nb = neigh[vc * KNB + j];
        g[j] = pool[(size_t)batch * NV + nb];
    }
    g[7] = make_float2(0.f, 0.f);

    // W as 8 uniform pairs (pair 7 = padding).
    float2 wv[8];
    #pragma unroll
    for (int j = 0; j < KNB; ++j) wv[j] = make_float2(W[2 * j], W[2 * j + 1]);
    wv[7] = make_float2(0.f, 0.f);

    const bool col0 = (row == 0);          // lanes 0 and 16 carry B column 0

    v8f acc = (v8f){0.f, 0.f, 0.f, 0.f, 0.f, 0.f, 0.f, 0.f};
    #pragma unroll
    for (int s = 0; s < 4; ++s) {
        float2 glo = g[2 * s], ghi = g[2 * s + 1];
        float2 wlo = wv[2 * s], whi = wv[2 * s + 1];
        v2f a;  a.x  = hi ? ghi.x : glo.x;
                a.y  = hi ? ghi.y : glo.y;
        v2f bb; bb.x = col0 ? (hi ? whi.x : wlo.x) : 0.f;
                bb.y = col0 ? (hi ? whi.y : wlo.y) : 0.f;
        acc = __builtin_amdgcn_wmma_f32_16x16x4_f32(
                  /*neg_a=*/false, a, /*neg_b=*/false, bb,
                  /*c_mod=*/(short)0, acc, /*reuse_a=*/false, /*reuse_b=*/false);
    }

    float bs = bias[0];
    if (col0) {                            // lanes 0 and 16 hold D column 0
        int base = tile * 16 + (hi ? 8 : 0);
        #pragma unroll
        for (int r = 0; r < 8; ++r) {
            int vv = base + r;
            if (vv < NV) {
                float z = acc[r] + bs;
                out[(size_t)batch * NV + vv] = 1.0f / (1.0f + expf(-z));
            }
        }
    }
}

extern "C" void kernel_launch(void* const* d_in, const int* in_sizes, int n_in,
                              void* d_out, int out_size, void* d_ws, size_t ws_size,
                              hipStream_t stream) {
    const float* x     = (const float*)d_in[0];   // (16, 256, 40962) f32
    const int*   neigh = (const int*)  d_in[1];   // (40962*7,) i32
    const float* W     = (const float*)d_in[2];   // (1, 14) f32
    const float* bias  = (const float*)d_in[3];   // (1,) f32
    float* out   = (float*)d_out;                 // (16, 1, 40962) f32 flat
    float2* pool = (float2*)d_ws;                 // 16*40962 float2 = 5.24 MB

    // Kernel 1: streaming channel reduction (bandwidth-bound, ~29us floor).
    {
        int total  = NBATCH * NVH;                // 327,696 threads
        int blocks = (total + 255) / 256;         // 1281
        pool_reduce_kernel<<<blocks, 256, 0, stream>>>(x, pool);
    }
    // Kernel 2: gather + WMMA einsum + sigmoid. One wave per 16-vertex tile.
    {
        int waves   = NBATCH * TILES_PER_BATCH;   // 40,976 waves
        int threads = waves * 32;                 // 1,311,232
        int blocks  = (threads + 255) / 256;      // 5122
        attention_wmma_kernel<<<blocks, 256, 0, stream>>>(pool, neigh, W, bias, out);
    }
}